// Layer_65919158059102
// MI455X (gfx1250) — compile-verified
//
#include <hip/hip_runtime.h>

// B = 1048576 rows, DIM = 16. out = cos(phase)*x + sin(phase)*Jx with
// x = [p,q] - r, phase = -2*h*alpha*eps^2 * exp(-eps^2 * ||x||^2).
//
// Memory-bound: 256 MiB total traffic -> ~11.5 us floor at 23.3 TB/s.
// Strategy: one thread per float4 chunk (4 lanes per row) => perfectly
// coalesced global_load_b128/global_store_b128 with non-temporal (TH_NT)
// cache hints; per-row ||x||^2 via two wave32 xor-shuffles.

typedef float v4f __attribute__((ext_vector_type(4)));

__global__ __launch_bounds__(256) void symp_rot_kernel(
    const float* __restrict__ p,
    const float* __restrict__ q,
    const float* __restrict__ h,
    const float* __restrict__ r,
    const float* __restrict__ eps,
    const float* __restrict__ alpha,
    float* __restrict__ outp,
    float* __restrict__ outq,
    int total_chunks)           // B * 4
{
    const int g = blockIdx.x * blockDim.x + threadIdx.x;  // float4 index
    if (g >= total_chunks) return;
    const int chunk = g & 3;                              // 0..3 within row

    // Uniform scalars: compiler scalarizes to s_load + SALU.
    const float e2 = eps[0] * eps[0];
    const float k  = -2.0f * h[0] * alpha[0] * e2;

    const v4f* p4 = (const v4f*)p;
    const v4f* q4 = (const v4f*)q;
    const v4f* r4 = (const v4f*)r;                        // r[0:16] then r[16:32]

    // Streaming loads: 16 B per lane, contiguous across the wave.
    v4f xp = __builtin_nontemporal_load(&p4[g]) - r4[chunk];
    v4f xq = __builtin_nontemporal_load(&q4[g]) - r4[4 + chunk];

    // Partial squared norm for this lane's 8 elements.
    float part = xp.x * xp.x + xp.y * xp.y + xp.z * xp.z + xp.w * xp.w
               + xq.x * xq.x + xq.y * xq.y + xq.z * xq.z + xq.w * xq.w;

    // Row reduction across the 4 lanes owning this row (wave32 cross-lane).
    part += __shfl_xor(part, 1);
    part += __shfl_xor(part, 2);   // now full ||x||^2 in all 4 lanes

    // phase magnitude is modest -> native transcendentals are plenty accurate.
    const float phase = k * __expf(-e2 * part);
    const float c = __cosf(phase);
    const float s = __sinf(phase);

    // out_p = c*xp + s*xq ; out_q = c*xq - s*xp   (J@x = [xq, -xp])
    v4f op = c * xp + s * xq;
    v4f oq = c * xq - s * xp;

    __builtin_nontemporal_store(op, (v4f*)outp + g);
    __builtin_nontemporal_store(oq, (v4f*)outq + g);
}

extern "C" void kernel_launch(void* const* d_in, const int* in_sizes, int n_in,
                              void* d_out, int out_size, void* d_ws, size_t ws_size,
                              hipStream_t stream) {
    // setup_inputs() order: p, q, h, r, epsilon, alpha
    const float* p     = (const float*)d_in[0];
    const float* q     = (const float*)d_in[1];
    const float* h     = (const float*)d_in[2];
    const float* r     = (const float*)d_in[3];
    const float* eps   = (const float*)d_in[4];
    const float* alpha = (const float*)d_in[5];

    const int B = in_sizes[0] / 16;              // p is [B,16]
    float* outp = (float*)d_out;                 // first half: rotated p
    float* outq = (float*)d_out + (size_t)B * 16; // second half: rotated q

    const int total_chunks = B * 4;              // one thread per float4
    const int block = 256;
    const int grid  = (total_chunks + block - 1) / block;

    symp_rot_kernel<<<grid, block, 0, stream>>>(p, q, h, r, eps, alpha,
                                                outp, outq, total_chunks);
}